// PolicyGradientLoss_2001454760826
// MI455X (gfx1250) — compile-verified
//
#include <hip/hip_runtime.h>
#include <stdint.h>

#define PG_B 64
#define PG_S 512
#define PG_A 4096

// ---------------------------------------------------------------------------
// Kernel 1: one workgroup per (b,s) row.
//   - async-DMA the 4096-float row from global into LDS (CDNA5 async path)
//   - block reduce max, then sum(exp(x - max))
//   - thread 0 gathers row[action], applies terminal mask, writes to ws
// ---------------------------------------------------------------------------
__global__ __launch_bounds__(256) void pg_row_kernel(const int* __restrict__ actions,
                                                     const float* __restrict__ logits,
                                                     const int* __restrict__ terminals,
                                                     float* __restrict__ chosen_ws)
{
    __shared__ __align__(16) float row[PG_A];   // 16 KB row tile
    __shared__ float red[256];

    const int r   = blockIdx.x;                 // r = b*S + s
    const int tid = threadIdx.x;
    const float* src = logits + (size_t)r * PG_A;

    // 16 KB = 4 chunks of (256 lanes x 16B). Per-lane async b128 global->LDS.
    // Low 32 bits of a generic pointer to __shared__ are the wave-relative
    // LDS byte address (aperture rule), which is exactly what VDST wants.
#pragma unroll
    for (int c = 0; c < 4; ++c) {
        const int elem = c * 1024 + tid * 4;
        unsigned lds_addr = (unsigned)(unsigned long long)(&row[elem]);
        const float* g = src + elem;
        asm volatile("global_load_async_to_lds_b128 %0, %1, off"
                     :: "v"(lds_addr), "v"(g)
                     : "memory");
    }
    asm volatile("s_wait_asynccnt 0" ::: "memory");
    __syncthreads();   // every lane reads data moved by other waves

    // ---- pass 1: row max ----
    float m = -INFINITY;
#pragma unroll
    for (int i = tid; i < PG_A; i += 256) m = fmaxf(m, row[i]);
    red[tid] = m;
    __syncthreads();
    for (int off = 128; off > 0; off >>= 1) {
        if (tid < off) red[tid] = fmaxf(red[tid], red[tid + off]);
        __syncthreads();
    }
    const float rowmax = red[0];
    __syncthreads();

    // ---- pass 2: sum(exp(x - max)) ----
    float ssum = 0.0f;
#pragma unroll
    for (int i = tid; i < PG_A; i += 256) ssum += __expf(row[i] - rowmax);
    red[tid] = ssum;
    __syncthreads();
    for (int off = 128; off > 0; off >>= 1) {
        if (tid < off) red[tid] += red[tid + off];
        __syncthreads();
    }

    if (tid == 0) {
        const int   act    = actions[r];
        const float chosen = row[act];
        const float logp   = chosen - rowmax - __logf(red[0]);
        const float mask   = 1.0f - (float)terminals[r];
        chosen_ws[r] = logp * mask;
    }
}

// ---------------------------------------------------------------------------
// Kernel 2: per-batch deterministic tree reductions.
//   partial[b] = (sum_s chosen[b,s]) * (sum_s rewards[b,s])
// ---------------------------------------------------------------------------
__global__ __launch_bounds__(512) void pg_b_kernel(const float* __restrict__ chosen_ws,
                                                   const float* __restrict__ rewards,
                                                   float* __restrict__ partial)
{
    __shared__ float red[512];
    const int b   = blockIdx.x;
    const int tid = threadIdx.x;

    red[tid] = chosen_ws[b * PG_S + tid];
    __syncthreads();
    for (int off = 256; off > 0; off >>= 1) {
        if (tid < off) red[tid] += red[tid + off];
        __syncthreads();
    }
    const float cs = red[0];
    __syncthreads();

    red[tid] = rewards[b * PG_S + tid];
    __syncthreads();
    for (int off = 256; off > 0; off >>= 1) {
        if (tid < off) red[tid] += red[tid + off];
        __syncthreads();
    }
    if (tid == 0) partial[b] = cs * red[0];
}

// ---------------------------------------------------------------------------
// Kernel 3: final mean:  out = -(1/B) * sum_b partial[b]
// ---------------------------------------------------------------------------
__global__ __launch_bounds__(64) void pg_final_kernel(const float* __restrict__ partial,
                                                      float* __restrict__ out)
{
    __shared__ float red[64];
    const int tid = threadIdx.x;
    red[tid] = partial[tid];
    __syncthreads();
    for (int off = 32; off > 0; off >>= 1) {
        if (tid < off) red[tid] += red[tid + off];
        __syncthreads();
    }
    if (tid == 0) out[0] = -red[0] / (float)PG_B;
}

extern "C" void kernel_launch(void* const* d_in, const int* in_sizes, int n_in,
                              void* d_out, int out_size, void* d_ws, size_t ws_size,
                              hipStream_t stream)
{
    const int*   actions   = (const int*)d_in[0];    // [B,S] int32
    const float* logits    = (const float*)d_in[1];  // [B,S,A] f32
    const float* rewards   = (const float*)d_in[2];  // [B,S] f32
    const int*   terminals = (const int*)d_in[3];    // [B,S] int32
    float* out = (float*)d_out;                      // scalar f32

    float* chosen_ws = (float*)d_ws;                 // B*S floats (128 KB)
    float* partial   = chosen_ws + PG_B * PG_S;      // B floats

    pg_row_kernel<<<PG_B * PG_S, 256, 0, stream>>>(actions, logits, terminals, chosen_ws);
    pg_b_kernel<<<PG_B, 512, 0, stream>>>(chosen_ws, rewards, partial);
    pg_final_kernel<<<1, 64, 0, stream>>>(partial, out);
}